// SpectralEMA_16716012716504
// MI455X (gfx1250) — compile-verified
//
#include <hip/hip_runtime.h>
#include <math.h>

// SpectralEMA scan — memory-bound streaming kernel for MI455X (gfx1250).
//
// Math: state*exp(i(angle(x)-angle(state))) = |state| * x/|x|, so
//   new = x * (a*|state|/|x| + (1-rho)),
// and only m=|state| is loop-carried:
//   r = |x_t| ;  z = a*m + (1-rho)*r ;  m' = |z|
// Final output = (x_S/|x_S|) * z_S.
//
// Roofline: 512 MiB one-touch stream / 23.3 TB/s ~= 22.5 us; compute ~1 GFLOP
// + 128M sqrt -> pure HBM bound. Hence: NT load hints (stream >> 192 MB L2),
// full-coverage global_prefetch 32 steps ahead (steps are 16 KB apart, one
// cacheline set each), 8-step unroll for 16 in-flight B64 loads per wave.

typedef float v2f __attribute__((ext_vector_type(2)));
typedef float v4f __attribute__((ext_vector_type(4)));

#define PI_F 3.14159265358979323846f

__global__ __launch_bounds__(256) void spectral_ema_scan(
    const float* __restrict__ fft_real,
    const float* __restrict__ fft_imag,
    const float* __restrict__ init_real,
    const float* __restrict__ init_imag,
    const float* __restrict__ rho_logit,
    const float* __restrict__ theta_raw,
    float* __restrict__ out,
    int S, int F)
{
    // 2D grid: y = batch, x tiles the F/2 feature pairs (no div/mod, no guard
    // needed when F/2 is a multiple of blockDim.x; launch ensures coverage).
    const int pair = blockIdx.x * blockDim.x + threadIdx.x;
    const int b    = blockIdx.y;
    const int f0   = pair << 1;                 // 2 features per thread
    if (f0 >= F) return;

    // Per-feature decay parameters (once per 512-step scan; negligible).
    float ar[2], ai[2], omr[2], m[2];
    #pragma unroll
    for (int j = 0; j < 2; ++j) {
        const float rl  = rho_logit[f0 + j];
        const float rho = 1.0f / (1.0f + expf(-rl));
        const float th  = PI_F * tanhf(theta_raw[f0 + j]);
        float sn, cs;
        sincosf(th, &sn, &cs);
        ar[j]  = rho * cs;
        ai[j]  = rho * sn;
        omr[j] = 1.0f - rho;
        const float ir = init_real[(size_t)b * F + f0 + j];
        const float ii = init_imag[(size_t)b * F + f0 + j];
        m[j] = sqrtf(ir * ir + ii * ii);
    }

    const size_t base    = (size_t)b * S * F + f0;
    const v2f*   xr_p    = (const v2f*)(fft_real + base);
    const v2f*   xi_p    = (const v2f*)(fft_imag + base);
    const int    strideV = F >> 1;              // v2f elements per time step

    float zr[2] = {0.0f, 0.0f}, zi[2] = {0.0f, 0.0f};
    float last_r[2] = {0.0f, 0.0f};
    v2f   lxr = {0.0f, 0.0f}, lxi = {0.0f, 0.0f};

    constexpr int UNROLL = 8;                   // 16 B64 loads in flight/wave
    constexpr int PFD    = 32;                  // prefetch distance (steps)

    for (int s = 0; s < S; s += UNROLL) {
        // Full-coverage prefetch: every step of the block PFD ahead (each step
        // is its own cacheline set, 16 KB apart). One guard per block.
        if (s + PFD + UNROLL <= S) {
            #pragma unroll
            for (int u = 0; u < UNROLL; ++u) {
                const size_t pofs = (size_t)(s + u + PFD) * strideV;
                __builtin_prefetch((const void*)(xr_p + pofs), 0, 1);
                __builtin_prefetch((const void*)(xi_p + pofs), 0, 1);
            }
        }
        #pragma unroll
        for (int u = 0; u < UNROLL; ++u) {
            // One-touch stream: non-temporal so we don't thrash the 192MB L2.
            const v2f xr = __builtin_nontemporal_load(xr_p + (size_t)(s + u) * strideV);
            const v2f xi = __builtin_nontemporal_load(xi_p + (size_t)(s + u) * strideV);
            #pragma unroll
            for (int j = 0; j < 2; ++j) {
                const float r = sqrtf(xr[j] * xr[j] + xi[j] * xi[j]);
                zr[j] = ar[j] * m[j] + omr[j] * r;    // z = a*m + (1-rho)*r
                zi[j] = ai[j] * m[j];
                m[j]  = sqrtf(zr[j] * zr[j] + zi[j] * zi[j]);
                last_r[j] = r;
            }
            lxr = xr;
            lxi = xi;
        }
    }

    // out = (x_S / |x_S|) * z_S ;  angle(0)=0 => unit direction (1,0) if r==0.
    v4f o;
    #pragma unroll
    for (int j = 0; j < 2; ++j) {
        float outr, outi;
        if (last_r[j] > 0.0f) {
            const float inv = 1.0f / last_r[j];
            outr = (lxr[j] * zr[j] - lxi[j] * zi[j]) * inv;
            outi = (lxr[j] * zi[j] + lxi[j] * zr[j]) * inv;
        } else {
            outr = zr[j];
            outi = zi[j];
        }
        o[2 * j + 0] = outr;
        o[2 * j + 1] = outi;
    }
    *(v4f*)(out + ((size_t)b * F + f0) * 2) = o;    // [B,F,2], 16B contiguous
}

extern "C" void kernel_launch(void* const* d_in, const int* in_sizes, int n_in,
                              void* d_out, int out_size, void* d_ws, size_t ws_size,
                              hipStream_t stream) {
    (void)n_in; (void)out_size; (void)d_ws; (void)ws_size;

    const float* fft_real  = (const float*)d_in[0];
    const float* fft_imag  = (const float*)d_in[1];
    const float* init_real = (const float*)d_in[2];
    const float* init_imag = (const float*)d_in[3];
    const float* rho_logit = (const float*)d_in[4];
    const float* theta_raw = (const float*)d_in[5];
    float* out = (float*)d_out;

    const int F = in_sizes[4];                  // rho_logit is [F]
    const int B = in_sizes[2] / F;              // init_real is [B,F]
    const int S = in_sizes[0] / (B * F);        // fft_real is [B,S,F]

    const int block  = 256;                     // 8 wave32 per workgroup
    const int pairs  = F / 2;
    dim3 grid((pairs + block - 1) / block, B, 1);
    spectral_ema_scan<<<grid, dim3(block, 1, 1), 0, stream>>>(
        fft_real, fft_imag, init_real, init_imag, rho_logit, theta_raw,
        out, S, F);
}